// GCN_19344532702046
// MI455X (gfx1250) — compile-verified
//
#include <hip/hip_runtime.h>

// ---------------------------------------------------------------------------
// GCN forward on gfx1250 (MI455X):
//   h1 = x @ W1 + b1                      (fp32 WMMA 16x16x4, exact precision)
//   a1 = spmm(edges, h1)                  (f32 hardware atomics, L2-resident)
//   h2 = elu(a1) @ W2 + b2                (ELU fused into A-load)
//   a2 = spmm(edges, h2)
//   out = elu(a2) @ Wo + bo
// ---------------------------------------------------------------------------

typedef float v2f __attribute__((ext_vector_type(2)));
typedef float v8f __attribute__((ext_vector_type(8)));

#define GCN_NODES 50000
#define GCN_NFEAT 128
#define GCN_NHID  128
#define GCN_NOUT  64

__device__ __forceinline__ float gcn_elu(float x) {
    return x > 0.0f ? x : (__expf(x) - 1.0f);
}

// ---------------------------------------------------------------------------
// GEMM: C[M x NCOLS] = op(A)[M x K] * W[K x NCOLS] + bias, op = identity | elu
// One wave computes a 16-row x NCOLS band: A fragment loaded once per k-step
// (single contiguous float2 per lane), reused across NCOLS/16 B tiles.
// 4 waves / block. M must be a multiple of 16 (50000 = 16 * 3125).
// ---------------------------------------------------------------------------
template <int NCOLS, bool ELU_A>
__global__ __launch_bounds__(128) void gcn_gemm_wmma(
        const float* __restrict__ A, const float* __restrict__ W,
        const float* __restrict__ bias, float* __restrict__ C,
        int M, int K) {
    constexpr int NT = NCOLS / 16;
    const int lane  = threadIdx.x & 31;
    const int wave  = threadIdx.x >> 5;
    const int mtile = blockIdx.x * 4 + wave;
    const int m0    = mtile * 16;
    if (m0 >= M) return;                 // wave-uniform: EXEC stays all-1s

    const int half = lane >> 4;          // 0: lanes 0-15, 1: lanes 16-31
    const int lr   = lane & 15;

    v8f c[NT];
#pragma unroll
    for (int t = 0; t < NT; ++t) c[t] = v8f{0, 0, 0, 0, 0, 0, 0, 0};

    const float* arow = A + (size_t)(m0 + lr) * K;

    for (int kk = 0; kk < K; kk += 4) {
        const int ka = kk + 2 * half;    // this lane's K pair
        v2f a;
        a.x = arow[ka];                  // contiguous pair -> b64 load
        a.y = arow[ka + 1];
        if (ELU_A) { a.x = gcn_elu(a.x); a.y = gcn_elu(a.y); }
#pragma unroll
        for (int t = 0; t < NT; ++t) {
            v2f b;
            b.x = W[(size_t)ka * NCOLS + t * 16 + lr];
            b.y = W[(size_t)(ka + 1) * NCOLS + t * 16 + lr];
            c[t] = __builtin_amdgcn_wmma_f32_16x16x4_f32(
                false, a, false, b, (short)0, c[t], false, false);
        }
    }

#pragma unroll
    for (int t = 0; t < NT; ++t) {
        const int col = t * 16 + lr;
        const float bb = bias[col];
#pragma unroll
        for (int r = 0; r < 8; ++r) {
            const int row = m0 + r + 8 * half;     // C layout: VGPR r -> M = r + 8*half
            C[(size_t)row * NCOLS + col] = c[t][r] + bb;
        }
    }
}

// ---------------------------------------------------------------------------
// Zero-fill (float4 vectorized)
// ---------------------------------------------------------------------------
__global__ __launch_bounds__(256) void gcn_zero(float4* __restrict__ p, size_t n4) {
    size_t i = (size_t)blockIdx.x * blockDim.x + threadIdx.x;
    if (i < n4) p[i] = float4{0.0f, 0.0f, 0.0f, 0.0f};
}

// ---------------------------------------------------------------------------
// COO SpMM scatter: acc[row] += val * h[col].
// One wave per edge: 32 lanes cover the 128-feature row as float4 chunks
// (coalesced 512B gather of h[col], L2-resident), 4 hw f32 atomics per lane.
// Edge index is wave-uniform -> force to SGPR via readfirstlane so the edge
// metadata (row/col/val) lowers to scalar s_load instead of 32 redundant
// per-lane vector loads.
// ---------------------------------------------------------------------------
__global__ __launch_bounds__(256) void gcn_spmm_scatter(
        const int* __restrict__ erow, const int* __restrict__ ecol,
        const float* __restrict__ eval, const float* __restrict__ h,
        float* __restrict__ acc, int E) {
    const long long tid = (long long)blockIdx.x * blockDim.x + threadIdx.x;
    const int e = __builtin_amdgcn_readfirstlane((int)(tid >> 5));  // wave-uniform edge id
    if (e >= E) return;
    const int c4 = (threadIdx.x & 31) * 4;   // this lane's feature chunk

    const int   r = erow[e];                 // uniform address -> s_load
    const int   c = ecol[e];
    const float v = eval[e];

    const float4 hv = *(const float4*)(h + (size_t)c * GCN_NHID + c4);
    float* p = acc + (size_t)r * GCN_NHID + c4;
    unsafeAtomicAdd(p + 0, v * hv.x);        // -> global_atomic_add_f32
    unsafeAtomicAdd(p + 1, v * hv.y);
    unsafeAtomicAdd(p + 2, v * hv.z);
    unsafeAtomicAdd(p + 3, v * hv.w);
}

// ---------------------------------------------------------------------------
extern "C" void kernel_launch(void* const* d_in, const int* in_sizes, int n_in,
                              void* d_out, int out_size, void* d_ws, size_t ws_size,
                              hipStream_t stream) {
    const float* x    = (const float*)d_in[0];
    const int*   erow = (const int*)  d_in[1];
    const int*   ecol = (const int*)  d_in[2];
    const float* evl  = (const float*)d_in[3];
    const float* W1   = (const float*)d_in[4];
    const float* b1   = (const float*)d_in[5];
    const float* W2   = (const float*)d_in[6];
    const float* b2   = (const float*)d_in[7];
    const float* Wo   = (const float*)d_in[8];
    const float* bo   = (const float*)d_in[9];
    const int E = in_sizes[1];

    float* out    = (float*)d_out;
    float* bufH   = (float*)d_ws;                              // 50000*128 f32
    float* bufAcc = bufH + (size_t)GCN_NODES * GCN_NHID;       // 50000*128 f32

    const int mtiles = GCN_NODES / 16;                         // 3125
    const dim3 gblk(128);
    const dim3 ggrd((mtiles + 3) / 4);                         // 4 waves/block

    const size_t n4 = (size_t)GCN_NODES * GCN_NHID / 4;
    const dim3 zgrd((unsigned)((n4 + 255) / 256));

    const long long sthreads = (long long)E * (GCN_NHID / 4);
    const dim3 sgrd((unsigned)((sthreads + 255) / 256));

    // Layer 1: h = x @ W1 + b1
    gcn_gemm_wmma<GCN_NHID, false><<<ggrd, gblk, 0, stream>>>(x, W1, b1, bufH,
                                                              GCN_NODES, GCN_NFEAT);
    // acc = spmm(edges, h)
    gcn_zero<<<zgrd, 256, 0, stream>>>((float4*)bufAcc, n4);
    gcn_spmm_scatter<<<sgrd, 256, 0, stream>>>(erow, ecol, evl, bufH, bufAcc, E);

    // Layer 2: h = elu(acc) @ W2 + b2   (ELU fused into A-load)
    gcn_gemm_wmma<GCN_NHID, true><<<ggrd, gblk, 0, stream>>>(bufAcc, W2, b2, bufH,
                                                             GCN_NODES, GCN_NHID);
    // acc = spmm(edges, h)
    gcn_zero<<<zgrd, 256, 0, stream>>>((float4*)bufAcc, n4);
    gcn_spmm_scatter<<<sgrd, 256, 0, stream>>>(erow, ecol, evl, bufH, bufAcc, E);

    // Output layer: out = elu(acc) @ Wo + bo
    gcn_gemm_wmma<GCN_NOUT, true><<<ggrd, gblk, 0, stream>>>(bufAcc, Wo, bo, out,
                                                             GCN_NODES, GCN_NHID);
}